// LatentTree_34969623724736
// MI455X (gfx1250) — compile-verified
//
#include <hip/hip_runtime.h>

typedef __attribute__((ext_vector_type(2))) float v2f;
typedef __attribute__((ext_vector_type(8))) float v8f;

#define DIM       128
#define NB_SPLIT  1023
#define NB_NODES  2047
#define ROWS      16      // rows of x per block
#define STRIDE    2052    // LDS row stride (padded: breaks 8-row bank aliasing)
#define THREADS   256     // 8 wave32

__global__ __launch_bounds__(THREADS)
void latent_tree_kernel(const float* __restrict__ x,
                        const float* __restrict__ A,
                        float* __restrict__ out)
{
    __shared__ float m[ROWS * STRIDE];   // ~128 KB of the 320 KB WGP LDS

    const int tid  = threadIdx.x;
    const int lane = tid & 31;
    const int wave = tid >> 5;
    const int row0 = blockIdx.x * ROWS;

    // V_WMMA_F32_16X16X4_F32 operand layout (wave32):
    //  A (16x4): lane L holds M = L%16, K = k0 + 2*(L/16) + {0,1}
    //  B (4x16): lane L holds N = L%16, K = k0 + 2*(L/16) + {0,1}
    //  D (16x16): VGPR r, lane L -> row = r + 8*(L/16), col = L%16
    const int laneM = lane & 15;
    const int laneK = (lane >> 4) << 1;   // 0 or 2

    // Preload this wave's x fragments for all K (kept in 64 VGPRs).
    const float* xp = x + (size_t)(row0 + laneM) * DIM + laneK;
    v2f ax[32];
#pragma unroll
    for (int k = 0; k < 32; ++k)
        ax[k] = *(const v2f*)(xp + 4 * k);

    // Each wave computes 8 column tiles of 16 splits: tiles wave*8 .. wave*8+7.
    for (int t = 0; t < 8; ++t) {
        const int col0 = (wave * 8 + t) * 16;
        const int col  = col0 + laneM;            // split id this lane produces
        const int colc = col > (NB_SPLIT - 1) ? (NB_SPLIT - 1) : col; // clamp pad col
        const float* ap = A + (size_t)colc * DIM + laneK;

        v8f acc = {};
#pragma unroll
        for (int k = 0; k < 32; ++k) {
            v2f b = *(const v2f*)(ap + 4 * k);
            acc = __builtin_amdgcn_wmma_f32_16x16x4_f32(
                      /*neg_a=*/false, ax[k], /*neg_b=*/false, b,
                      /*c_mod=*/(short)0, acc,
                      /*reuse_a=*/false, /*reuse_b=*/false);
        }

        // Scatter edge values straight into the children slots of the tree.
        if (col < NB_SPLIT) {
            const int cbase = 2 * col + 1;
            const int rhalf = (lane >> 4) << 3;   // 0 or 8
#pragma unroll
            for (int r = 0; r < 8; ++r) {
                const float v = acc[r];
                float* mr = m + (r + rhalf) * STRIDE;
                mr[cbase]     = v;     // left child edge:  XA[s]
                mr[cbase + 1] = -v;    // right child edge: -XA[s]
            }
        }
    }
    __syncthreads();

    // Root value.
    if (tid < ROWS) m[tid * STRIDE] = 1.0f;
    __syncthreads();

    // Top-down path-min propagation, level by level.
    for (int d = 0; d < 10; ++d) {
        const int lvl  = 1 << d;        // nodes at this level
        const int base = lvl - 1;       // first node id at this level
        for (int i = tid; i < ROWS * lvl; i += THREADS) {
            const int row = i >> d;
            const int off = i & (lvl - 1);
            const int s   = base + off;
            float* mr = m + row * STRIDE;
            const float p = mr[s];
            const int  c  = 2 * s + 1;
            mr[c]     = fminf(mr[c],     p);
            mr[c + 1] = fminf(mr[c + 1], p);
        }
        __syncthreads();
    }

    // Clamp to [0,1] and store (contiguous per row -> coalesced b32 stores).
    for (int row = 0; row < ROWS; ++row) {
        const size_t ob = (size_t)(row0 + row) * NB_NODES;
        const float* mr = m + row * STRIDE;
        for (int j = tid; j < NB_NODES; j += THREADS) {
            float v = mr[j];
            v = v < 0.0f ? 0.0f : (v > 1.0f ? 1.0f : v);
            out[ob + j] = v;
        }
    }
}

extern "C" void kernel_launch(void* const* d_in, const int* in_sizes, int n_in,
                              void* d_out, int out_size, void* d_ws, size_t ws_size,
                              hipStream_t stream) {
    const float* x = (const float*)d_in[0];   // [N, 128] f32
    const float* A = (const float*)d_in[1];   // [1023, 128] f32
    float* out = (float*)d_out;               // [N, 2047] f32

    const int n_rows = in_sizes[0] / DIM;     // 65536
    dim3 grid(n_rows / ROWS);                 // 4096 blocks
    dim3 block(THREADS);
    latent_tree_kernel<<<grid, block, 0, stream>>>(x, A, out);
}